// EquivariantAttention_16415365006062
// MI455X (gfx1250) — compile-verified
//
#include <hip/hip_runtime.h>
#include <math.h>

typedef __attribute__((ext_vector_type(2))) float v2f;
typedef __attribute__((ext_vector_type(8))) float v8f;

#define NN   25000
#define EE   400000
#define DIMM 128
#define HH   8

// ---- ordered-uint encoding so float max can use GLOBAL_ATOMIC_MAX_U32 ----
__device__ __forceinline__ unsigned fenc(float f) {
  unsigned u = __float_as_uint(f);
  return (u & 0x80000000u) ? ~u : (u | 0x80000000u);
}
__device__ __forceinline__ float fdec(unsigned u) {
  return (u & 0x80000000u) ? __uint_as_float(u ^ 0x80000000u) : __uint_as_float(~u);
}

__global__ void zero_f32(float* p, long long n) {
  long long i = (long long)blockIdx.x * blockDim.x + threadIdx.x;
  if (i < n) p[i] = 0.0f;
}
__global__ void fill_u32(unsigned* p, long long n, unsigned v) {
  long long i = (long long)blockIdx.x * blockDim.x + threadIdx.x;
  if (i < n) p[i] = v;
}

// ---------------------------------------------------------------------------
// Strided fp32 GEMM, 2x2 register blocking: one wave computes a 32x32 tile of
// C = A @ B (+bias per column) with V_WMMA_F32_16X16X4_F32.
// Each A fragment feeds 2 column tiles, each B fragment feeds 2 row tiles
// -> 2 loads per WMMA (and A loads are b64 when AVEC, i.e. sAc==1).
// A element at A[row*sAr + k*sAc]; B row-major K x ldb; C at C[row*sCr + n*sCc]
// ---------------------------------------------------------------------------
template <bool AVEC>
__global__ __launch_bounds__(32)
void gemm32_wmma(const float* __restrict__ A, long long sAr, long long sAc,
                 const float* __restrict__ B, int ldb,
                 const float* __restrict__ bias,
                 float* __restrict__ C, long long sCr, long long sCc,
                 int M, int K) {
  const int lane = threadIdx.x;
  const int half = lane >> 4;       // 0: lanes 0-15, 1: lanes 16-31
  const int r    = lane & 15;
  const int row0 = blockIdx.x * 32;
  const int col0 = blockIdx.y * 32;
  const int ar0  = (row0 + r      < M) ? (row0 + r)      : (M - 1);
  const int ar1  = (row0 + 16 + r < M) ? (row0 + 16 + r) : (M - 1);

  v8f acc00, acc01, acc10, acc11;   // [row tile][col tile]
  const float bi0 = bias ? bias[col0 + r]      : 0.0f;
  const float bi1 = bias ? bias[col0 + 16 + r] : 0.0f;
#pragma unroll
  for (int g = 0; g < 8; ++g) {
    acc00[g] = bi0; acc01[g] = bi1;
    acc10[g] = bi0; acc11[g] = bi1;
  }

#pragma unroll 4
  for (int kk = 0; kk < K; kk += 4) {
    const int kb = kk + 2 * half;   // A/B 16x4 layout: lanes 0-15 -> K kk+0/1,
    v2f a0, a1, b0, b1;             //                  lanes 16-31 -> K kk+2/3
    if (AVEC) {                     // sAc == 1: contiguous pair -> b64 load
      const float2 t0 = *(const float2*)(A + (long long)ar0 * sAr + kb);
      const float2 t1 = *(const float2*)(A + (long long)ar1 * sAr + kb);
      a0[0] = t0.x; a0[1] = t0.y;
      a1[0] = t1.x; a1[1] = t1.y;
    } else {
      a0[0] = A[(long long)ar0 * sAr + (long long)(kb + 0) * sAc];
      a0[1] = A[(long long)ar0 * sAr + (long long)(kb + 1) * sAc];
      a1[0] = A[(long long)ar1 * sAr + (long long)(kb + 0) * sAc];
      a1[1] = A[(long long)ar1 * sAr + (long long)(kb + 1) * sAc];
    }
    b0[0] = B[(kb + 0) * ldb + col0 + r];
    b0[1] = B[(kb + 1) * ldb + col0 + r];
    b1[0] = B[(kb + 0) * ldb + col0 + 16 + r];
    b1[1] = B[(kb + 1) * ldb + col0 + 16 + r];
    acc00 = __builtin_amdgcn_wmma_f32_16x16x4_f32(false, a0, false, b0,
                                                  (short)0, acc00, false, false);
    acc01 = __builtin_amdgcn_wmma_f32_16x16x4_f32(false, a0, false, b1,
                                                  (short)0, acc01, false, false);
    acc10 = __builtin_amdgcn_wmma_f32_16x16x4_f32(false, a1, false, b0,
                                                  (short)0, acc10, false, false);
    acc11 = __builtin_amdgcn_wmma_f32_16x16x4_f32(false, a1, false, b1,
                                                  (short)0, acc11, false, false);
  }

#pragma unroll
  for (int g = 0; g < 8; ++g) {     // C/D layout: VGPR g -> M = g + 8*half
    const int ra = row0 + g + 8 * half;
    const int rb = ra + 16;
    const long long ca = (long long)(col0 + r) * sCc;
    const long long cb = (long long)(col0 + 16 + r) * sCc;
    if (ra < M) {
      C[(long long)ra * sCr + ca] = acc00[g];
      C[(long long)ra * sCr + cb] = acc01[g];
    }
    if (rb < M) {
      C[(long long)rb * sCr + ca] = acc10[g];
      C[(long long)rb * sCr + cb] = acc11[g];
    }
  }
}

// ---------------------------------------------------------------------------
// Fused edge-bias MLP: bias[e,h] = SiLU(edge_attr @ W1 + b1) @ W2 + b2
// One wave per 16 edges. Hidden 16x64 staged in LDS (C-layout -> A-layout).
// ---------------------------------------------------------------------------
__global__ __launch_bounds__(32)
void edge_mlp_wmma(const float* __restrict__ ea, const float* __restrict__ W1,
                   const float* __restrict__ b1, const float* __restrict__ W2,
                   const float* __restrict__ b2, float* __restrict__ ebias) {
  __shared__ float hid[16][64];
  const int lane = threadIdx.x;
  const int half = lane >> 4;
  const int r    = lane & 15;
  const long long e0 = (long long)blockIdx.x * 16;

  // Layer 1: 16x64 output = 4 column tiles, K = 64
  for (int t = 0; t < 4; ++t) {
    v8f acc;
    const float bi = b1[t * 16 + r];
#pragma unroll
    for (int g = 0; g < 8; ++g) acc[g] = bi;
#pragma unroll 4
    for (int kk = 0; kk < 64; kk += 4) {
      const int kb = kk + 2 * half;
      v2f a, b;
      const float2 t0 = *(const float2*)(ea + (e0 + r) * 64 + kb);
      a[0] = t0.x; a[1] = t0.y;
      b[0] = W1[(kb + 0) * 64 + t * 16 + r];
      b[1] = W1[(kb + 1) * 64 + t * 16 + r];
      acc = __builtin_amdgcn_wmma_f32_16x16x4_f32(false, a, false, b,
                                                  (short)0, acc, false, false);
    }
#pragma unroll
    for (int g = 0; g < 8; ++g) {
      const float x = acc[g];
      // SiLU with raw v_rcp_f32 (avoid IEEE divide chain)
      hid[g + 8 * half][t * 16 + r] =
          x * __builtin_amdgcn_rcpf(1.0f + __expf(-x));
    }
  }
  __syncthreads();

  // Layer 2: 16x8 output (single 16x16 tile, cols 8..15 dead), K = 64
  v8f acc2;
  const int cw = r & 7;            // clamp column for W2 (64x8)
  const float bi2 = b2[cw];
#pragma unroll
  for (int g = 0; g < 8; ++g) acc2[g] = bi2;
#pragma unroll 4
  for (int kk = 0; kk < 64; kk += 4) {
    const int kb = kk + 2 * half;
    v2f a, b;
    a[0] = hid[r][kb + 0];
    a[1] = hid[r][kb + 1];
    b[0] = W2[(kb + 0) * 8 + cw];
    b[1] = W2[(kb + 1) * 8 + cw];
    acc2 = __builtin_amdgcn_wmma_f32_16x16x4_f32(false, a, false, b,
                                                 (short)0, acc2, false, false);
  }
  if (r < 8) {
#pragma unroll
    for (int g = 0; g < 8; ++g)
      ebias[(e0 + g + 8 * half) * HH + r] = acc2[g];
  }
}

// ---- pass A: scores + segment max (atomic max via ordered uint) -----------
__global__ __launch_bounds__(256)
void score_max(const float* __restrict__ q, const float* __restrict__ k,
               const int* __restrict__ ei, const float* __restrict__ el,
               const float* __restrict__ eb, float* __restrict__ sc,
               unsigned* __restrict__ m) {
  long long idx = (long long)blockIdx.x * blockDim.x + threadIdx.x;
  if (idx >= (long long)EE * HH) return;
  const int e  = (int)(idx >> 3);
  const int hd = (int)(idx & 7);
  const int i  = ei[e];
  const int j  = ei[EE + e];
  const float4* qp = (const float4*)(q + (long long)j * DIMM + hd * 16);
  const float4* kp = (const float4*)(k + (long long)i * DIMM + hd * 16);
  float s = 0.0f;
#pragma unroll
  for (int t = 0; t < 4; ++t) {
    float4 a = qp[t], b = kp[t];
    s += a.x * b.x + a.y * b.y + a.z * b.z + a.w * b.w;
  }
  s = s * 0.25f - el[e] + eb[idx];            // 1/sqrt(16) = 0.25
  sc[idx] = s;
  atomicMax(&m[(long long)j * HH + hd], fenc(s));
}

// ---- pass B: exp(score - max) + segment sum -------------------------------
__global__ __launch_bounds__(256)
void exp_sum(const int* __restrict__ ei, float* __restrict__ sc,
             const unsigned* __restrict__ m, float* __restrict__ den) {
  long long idx = (long long)blockIdx.x * blockDim.x + threadIdx.x;
  if (idx >= (long long)EE * HH) return;
  const int e  = (int)(idx >> 3);
  const int hd = (int)(idx & 7);
  const int j  = ei[EE + e];
  const float ex = __expf(sc[idx] - fdec(m[(long long)j * HH + hd]));
  sc[idx] = ex;
  atomicAdd(&den[(long long)j * HH + hd], ex);
}

// ---- pass B2: sc -> alpha in place (one v_rcp per (edge,head)) ------------
__global__ __launch_bounds__(256)
void normalize_alpha(const int* __restrict__ ei, float* __restrict__ sc,
                     const float* __restrict__ den) {
  long long idx = (long long)blockIdx.x * blockDim.x + threadIdx.x;
  if (idx >= (long long)EE * HH) return;
  const int e  = (int)(idx >> 3);
  const int hd = (int)(idx & 7);
  const int j  = ei[EE + e];
  sc[idx] = sc[idx] * __builtin_amdgcn_rcpf(den[(long long)j * HH + hd]);
}

// ---- pass C: alpha-weighted scatter into h_agg / v_agg (no divides) -------
__global__ __launch_bounds__(256)
void scatter_agg(const int* __restrict__ ei, const float* __restrict__ alpha,
                 const float* __restrict__ vh, const float* __restrict__ vv,
                 float* __restrict__ hagg, float* __restrict__ vagg) {
  long long idx = (long long)blockIdx.x * blockDim.x + threadIdx.x;
  if (idx >= (long long)EE * DIMM) return;
  const int e  = (int)(idx >> 7);
  const int d  = (int)(idx & 127);
  const int hd = d >> 4;
  const int i  = ei[e];
  const int j  = ei[EE + e];
  const float a = alpha[(long long)e * HH + hd];
  atomicAdd(&hagg[(long long)j * DIMM + d], a * vh[(long long)i * DIMM + d]);
#pragma unroll
  for (int c = 0; c < 3; ++c)
    atomicAdd(&vagg[(long long)j * (DIMM * 3) + d * 3 + c],
              a * vv[(long long)i * (DIMM * 3) + d * 3 + c]);
}

// ---------------------------------------------------------------------------
extern "C" void kernel_launch(void* const* d_in, const int* in_sizes, int n_in,
                              void* d_out, int out_size, void* d_ws, size_t ws_size,
                              hipStream_t stream) {
  const float* h   = (const float*)d_in[0];
  const float* v   = (const float*)d_in[1];
  const int*   ei  = (const int*)d_in[2];
  const float* ea  = (const float*)d_in[3];
  const float* el  = (const float*)d_in[4];
  const float* Wq  = (const float*)d_in[5];
  const float* bq  = (const float*)d_in[6];
  const float* Wk  = (const float*)d_in[7];
  const float* bk  = (const float*)d_in[8];
  const float* Wvh = (const float*)d_in[9];
  const float* bvh = (const float*)d_in[10];
  const float* Wvv = (const float*)d_in[11];
  const float* Woh = (const float*)d_in[12];
  const float* Wov = (const float*)d_in[13];
  const float* w1  = (const float*)d_in[14];
  const float* b1  = (const float*)d_in[15];
  const float* w2  = (const float*)d_in[16];
  const float* b2  = (const float*)d_in[17];

  float* ws = (float*)d_ws;
  float*    q    = ws; ws += (size_t)NN * DIMM;
  float*    kbuf = ws; ws += (size_t)NN * DIMM;
  float*    vh   = ws; ws += (size_t)NN * DIMM;
  float*    vv   = ws; ws += (size_t)NN * DIMM * 3;
  float*    eb   = ws; ws += (size_t)EE * HH;
  float*    sc   = ws; ws += (size_t)EE * HH;
  unsigned* m    = (unsigned*)ws; ws += (size_t)NN * HH;
  float*    den  = ws; ws += (size_t)NN * HH;
  float*    hagg = ws; ws += (size_t)NN * DIMM;
  float*    vagg = ws; ws += (size_t)NN * DIMM * 3;

  float* dh = (float*)d_out;
  float* dv = dh + (size_t)NN * DIMM;

  // init accumulators (ws is poisoned; must clear every call)
  {
    long long n1 = (long long)NN * DIMM;
    long long n3 = (long long)NN * DIMM * 3;
    long long nh = (long long)NN * HH;
    zero_f32<<<(unsigned)((n1 + 255) / 256), 256, 0, stream>>>(hagg, n1);
    zero_f32<<<(unsigned)((n3 + 255) / 256), 256, 0, stream>>>(vagg, n3);
    zero_f32<<<(unsigned)((nh + 255) / 256), 256, 0, stream>>>(den, nh);
    fill_u32<<<(unsigned)((nh + 255) / 256), 256, 0, stream>>>(m, nh, 0x007FFFFFu); // enc(-inf)
  }

  // node projections (fp32 WMMA GEMMs, 32x32 tile per wave)
  dim3 gn((NN + 31) / 32, DIMM / 32);
  gemm32_wmma<true><<<gn, 32, 0, stream>>>(h, DIMM, 1, Wq,  DIMM, bq,  q,    DIMM, 1, NN, DIMM);
  gemm32_wmma<true><<<gn, 32, 0, stream>>>(h, DIMM, 1, Wk,  DIMM, bk,  kbuf, DIMM, 1, NN, DIMM);
  gemm32_wmma<true><<<gn, 32, 0, stream>>>(h, DIMM, 1, Wvh, DIMM, bvh, vh,   DIMM, 1, NN, DIMM);
  for (int c = 0; c < 3; ++c)   // vv[n,f,c] = sum_d v[n,d,c] * Wvv[d,f]
    gemm32_wmma<false><<<gn, 32, 0, stream>>>(v + c, DIMM * 3, 3, Wvv, DIMM, nullptr,
                                              vv + c, DIMM * 3, 3, NN, DIMM);

  // edge bias MLP (fused WMMA)
  edge_mlp_wmma<<<EE / 16, 32, 0, stream>>>(ea, w1, b1, w2, b2, eb);

  // segment softmax (edge-parallel passes)
  const long long tEH = (long long)EE * HH;
  const unsigned gEH = (unsigned)((tEH + 255) / 256);
  score_max      <<<gEH, 256, 0, stream>>>(q, kbuf, ei, el, eb, sc, m);
  exp_sum        <<<gEH, 256, 0, stream>>>(ei, sc, m, den);
  normalize_alpha<<<gEH, 256, 0, stream>>>(ei, sc, den);
  const long long tED = (long long)EE * DIMM;
  scatter_agg<<<(unsigned)((tED + 255) / 256), 256, 0, stream>>>(ei, sc, vh, vv,
                                                                 hagg, vagg);

  // output projections
  gemm32_wmma<true><<<gn, 32, 0, stream>>>(hagg, DIMM, 1, Woh, DIMM, nullptr, dh, DIMM, 1, NN, DIMM);
  for (int c = 0; c < 3; ++c)
    gemm32_wmma<false><<<gn, 32, 0, stream>>>(vagg + c, DIMM * 3, 3, Wov, DIMM, nullptr,
                                              dv + c, DIMM * 3, 3, NN, DIMM);
}